// MeanConv_31318901523154
// MI455X (gfx1250) — compile-verified
//
#include <hip/hip_runtime.h>

#ifndef __has_builtin
#define __has_builtin(x) 0
#endif

typedef __attribute__((ext_vector_type(2))) float v2f;
typedef __attribute__((ext_vector_type(8))) float v8f;
// Exact param type from hipcc diagnostic for the async-LDS builtin:
typedef __attribute__((__vector_size__(4 * sizeof(int)))) int vsi4;
typedef __attribute__((address_space(1))) vsi4 g_vsi4;   // global / "__device__"
typedef __attribute__((address_space(3))) vsi4 l_vsi4;   // LDS

#if __has_builtin(__builtin_amdgcn_global_load_async_to_lds_b128)
#define HAVE_ASYNC_LDS 1
#else
#define HAVE_ASYNC_LDS 0
#endif

#if __has_builtin(__builtin_amdgcn_s_wait_asynccnt)
#define WAIT_ASYNCCNT(n) __builtin_amdgcn_s_wait_asynccnt(n)
#else
#define WAIT_ASYNCCNT(n) asm volatile("s_wait_asynccnt " #n ::: "memory")
#endif

// Problem constants (from the reference)
#define C_NUSERS 16384
#define C_NITEMS 32768
#define C_EMBED  64
#define C_LAYER  64

#define WAVES_PB 4          // 128 threads / block
#define KCHUNK   64         // K columns staged per chunk
#define LSTRIDE  68         // 64 + 4 pad floats: bank-conflict-free, keeps 16B align
#define SLICE    (16 * LSTRIDE)

__global__ __launch_bounds__(128) void meanconv_wmma(
    const float* __restrict__ adj, const float* __restrict__ emb,
    const float* __restrict__ wgt, float* __restrict__ out)
{
  __shared__ float lds[WAVES_PB * 2 * SLICE];

  const int lane = threadIdx.x & 31;
  const int wave = threadIdx.x >> 5;
  const int hi   = lane >> 4;        // 0: K pair {0,1}, 1: K pair {2,3}
  const int lo   = lane & 15;
  const int m0   = (blockIdx.x * WAVES_PB + wave) * 16;

  float* buf0 = &lds[wave * 2 * SLICE];
  float* buf1 = buf0 + SLICE;

  v8f acc[4] = {v8f{0.f}, v8f{0.f}, v8f{0.f}, v8f{0.f}};
  float deg = 0.0f;

  // Stage one 16 x 64 fp32 tile of adj into an LDS slice.
  // 8 instructions/wave, each lane moves 16B (b128): 2 rows per instruction.
  auto issue = [&](float* dst, int kc) {
#pragma unroll
    for (int it = 0; it < 8; ++it) {
      const int r = it * 2 + hi;
      const float* g = adj + ((long)(m0 + r) * C_NITEMS + kc + lo * 4);
      float* l = dst + (r * LSTRIDE + lo * 4);
#if HAVE_ASYNC_LDS
      __builtin_amdgcn_global_load_async_to_lds_b128(
          (g_vsi4*)g, (l_vsi4*)l, 0, 0);
#else
      *(float4*)l = *(const float4*)g;
#endif
    }
  };

  constexpr int NCHUNK = C_NITEMS / KCHUNK;  // 512

#if HAVE_ASYNC_LDS
  issue(buf0, 0);  // prologue prefetch
#endif

  for (int ch = 0; ch < NCHUNK; ++ch) {
    float* cur = (ch & 1) ? buf1 : buf0;
    const int kc = ch * KCHUNK;

#if HAVE_ASYNC_LDS
    if (ch + 1 < NCHUNK) {
      issue((ch & 1) ? buf0 : buf1, kc + KCHUNK);  // prefetch next
      WAIT_ASYNCCNT(8);   // first 8 (current buffer) complete, next 8 in flight
    } else {
      WAIT_ASYNCCNT(0);
    }
#else
    issue(cur, kc);
#if __has_builtin(__builtin_amdgcn_wave_barrier)
    __builtin_amdgcn_wave_barrier();  // keep cross-lane LDS store->load order
#endif
#endif

#pragma unroll
    for (int ks = 0; ks < KCHUNK / 4; ++ks) {
      // A fragment (16x4 f32): lane holds row `lo`, K = 4*ks + 2*hi + {0,1}
      v2f a = *(const v2f*)(cur + lo * LSTRIDE + ks * 4 + 2 * hi);
      deg += a.x + a.y;  // fused row-sum (degree)

      // B fragments (4x16 f32) straight from global: item_emb is L2-resident
      const float* b0 = emb + (long)(kc + ks * 4 + 2 * hi) * C_EMBED + lo;
#pragma unroll
      for (int j = 0; j < 4; ++j) {
        v2f b;
        b.x = b0[j * 16];
        b.y = b0[C_EMBED + j * 16];
        acc[j] = __builtin_amdgcn_wmma_f32_16x16x4_f32(
            false, a, false, b, (short)0, acc[j], false, false);
      }
    }
  }

  // ---- Epilogue: mean-scale + fused agg @ mean_weight (K=64) ----
  // Combine the half-wave partial degree sums: lanes m and m+16 hold row m.
  deg += __shfl_xor(deg, 16, 32);

  float sc[8];
#pragma unroll
  for (int r = 0; r < 8; ++r) {
    // C/D layout: VGPR r of lane l is row (r + 8*hi); deg[m] lives in lane m.
    float d = __shfl(deg, r + 8 * hi, 32);
    sc[r] = 1.0f / (d + 1e-8f);
  }

  // Stage scaled agg tile (16x64) into LDS for re-read in A-matrix layout.
#pragma unroll
  for (int j = 0; j < 4; ++j)
#pragma unroll
    for (int r = 0; r < 8; ++r)
      buf0[(r + 8 * hi) * LSTRIDE + j * 16 + lo] = acc[j][r] * sc[r];

  __syncthreads();  // uniform across all waves; orders LDS writes vs reads

  v8f o[4] = {v8f{0.f}, v8f{0.f}, v8f{0.f}, v8f{0.f}};
#pragma unroll
  for (int kk = 0; kk < 16; ++kk) {
    v2f a2 = *(const v2f*)(buf0 + lo * LSTRIDE + kk * 4 + 2 * hi);
    const float* wr = wgt + (kk * 4 + 2 * hi) * C_LAYER + lo;
#pragma unroll
    for (int j = 0; j < 4; ++j) {
      v2f b2;
      b2.x = wr[j * 16];
      b2.y = wr[C_LAYER + j * 16];
      o[j] = __builtin_amdgcn_wmma_f32_16x16x4_f32(
          false, a2, false, b2, (short)0, o[j], false, false);
    }
  }

#pragma unroll
  for (int j = 0; j < 4; ++j)
#pragma unroll
    for (int r = 0; r < 8; ++r)
      out[(long)(m0 + r + 8 * hi) * C_LAYER + j * 16 + lo] = o[j][r];
}

extern "C" void kernel_launch(void* const* d_in, const int* in_sizes, int n_in,
                              void* d_out, int out_size, void* d_ws, size_t ws_size,
                              hipStream_t stream) {
  (void)in_sizes; (void)n_in; (void)out_size; (void)d_ws; (void)ws_size;
  const float* adj = (const float*)d_in[0];
  const float* emb = (const float*)d_in[1];
  const float* wgt = (const float*)d_in[2];
  float* out = (float*)d_out;

  // 16384 rows / (4 waves * 16 rows) = 256 blocks, 128 threads (4 wave32s)
  dim3 grid(C_NUSERS / (WAVES_PB * 16));
  meanconv_wmma<<<grid, WAVES_PB * 32, 0, stream>>>(adj, emb, wgt, out);
}